// SubspaceLinopFactory_63118839382143
// MI455X (gfx1250) — compile-verified
//
#include <hip/hip_runtime.h>

// Toeplitz normal operator for MI455X (gfx1250), compile-only target.
//   A=5 subspace coeffs, C=12 coils, 256x256 images, 2x-oversampled 512x512 grid.
// Pipeline (one in-place complex64 workspace of 60*512*512 = 120 MiB in d_ws;
// note 120 MiB buffer + 52 MiB kernels ~ fits the 192 MB L2):
//   modpad -> row FFT -> col FFT -> WMMA per-pixel contraction -> col iFFT
//   -> row iFFT -> crop + conj(mps) coil-sum.
// All normalization (ortho fwd 1/512 * OVERSAMP^2 4 * ortho inv 1/512) folded
// into the contraction as 1/65536.

#define A_DIM 5
#define C_DIM 12
#define H_DIM 256
#define W_DIM 256
#define HP 512
#define WP 512
#define NPIX (HP * WP)            // 262144 pixels on padded grid
#define NPLANES (C_DIM * A_DIM)   // 60 complex planes
#define PAD 128
#define TOTAL_SCALE (4.0f / (512.0f * 512.0f))  // = 1/65536

typedef __attribute__((ext_vector_type(2))) float v2f;
typedef __attribute__((ext_vector_type(8))) float v8f;

// CDNA5 async global<->LDS path (ASYNCcnt-tracked). Guarded so the host pass
// and toolchains without the builtins fall back to plain loads/stores.
// The builtins take b64 payloads typed as <2 x i32> pointers: global side in
// AS(1), LDS side in AS(3).
#if defined(__has_builtin)
#if __has_builtin(__builtin_amdgcn_global_load_async_to_lds_b64) && \
    __has_builtin(__builtin_amdgcn_s_wait_asynccnt)
#define HAVE_ASYNC_LDS 1
#endif
#if __has_builtin(__builtin_amdgcn_global_store_async_from_lds_b64) && \
    __has_builtin(__builtin_amdgcn_s_wait_asynccnt)
#define HAVE_ASYNC_LDS_ST 1
#endif
#endif

#if defined(HAVE_ASYNC_LDS) || defined(HAVE_ASYNC_LDS_ST)
typedef __attribute__((__vector_size__(2 * sizeof(int)))) int b64vec;
#define AS1P(p) ((__attribute__((address_space(1))) b64vec*)(p))
#define AS3P(p) ((__attribute__((address_space(3))) b64vec*)(p))
#endif

// ---------------------------------------------------------------------------
// 1) coil modulation + centered zero-pad:  buf[c,a,y,x] = mps[c]*x[a] (center)
// ---------------------------------------------------------------------------
__global__ void __launch_bounds__(256) modpad_kernel(
    const float* __restrict__ xr, const float* __restrict__ xi,
    const float* __restrict__ mr, const float* __restrict__ mi,
    float2* __restrict__ buf) {
  long i = (long)blockIdx.x * 256 + threadIdx.x;  // [c][a][y][x], 60*NPIX total
  int x = (int)(i & (WP - 1));
  int y = (int)((i >> 9) & (HP - 1));
  long pa = i >> 18;                 // plane = c*5 + a
  int a = (int)(pa % A_DIM);
  int c = (int)(pa / A_DIM);
  float2 v; v.x = 0.f; v.y = 0.f;
  int yy = y - PAD, xx = x - PAD;
  if ((unsigned)yy < (unsigned)H_DIM && (unsigned)xx < (unsigned)W_DIM) {
    long sm = ((long)c * H_DIM + yy) * W_DIM + xx;
    long sx = ((long)a * H_DIM + yy) * W_DIM + xx;
    float mrv = mr[sm], miv = mi[sm];
    float xrv = xr[sx], xiv = xi[sx];
    v.x = mrv * xrv - miv * xiv;     // mps * x (complex)
    v.y = mrv * xiv + miv * xrv;
  }
  buf[i] = v;
}

// ---------------------------------------------------------------------------
// 2) 512-point radix-2 DIT FFT of one line per block (rows or columns),
//    unnormalized; sign = -1 forward, +1 inverse. LDS-resident butterflies.
//    Line gather/scatter goes through the CDNA5 async global<->LDS engine
//    (per-lane LDS destination addressing) when available.
// ---------------------------------------------------------------------------
__global__ void __launch_bounds__(256) fft512_lines(float2* __restrict__ buf,
                                                    int isCol, float sign) {
  __shared__ float sre[512];
  __shared__ float sim[512];
  __shared__ float2 stage[512];
  int line = blockIdx.x;             // 60*512 lines
  long base; long stride;
  if (isCol) {
    long img = (long)(line >> 9);
    int x = line & 511;
    base = img * (long)NPIX + x;
    stride = WP;
  } else {
    base = (long)line * WP;
    stride = 1;
  }
  int t = threadIdx.x;               // 256 threads, 2 points each

  // ---- gather line into LDS ----
#ifdef HAVE_ASYNC_LDS
  for (int e = t; e < 512; e += 256) {
    __builtin_amdgcn_global_load_async_to_lds_b64(
        AS1P(buf + base + (long)e * stride), AS3P(&stage[e]), 0, 0);
  }
  __builtin_amdgcn_s_wait_asynccnt(0);
  __syncthreads();
  for (int e = t; e < 512; e += 256) {
    float2 v = stage[e];
    unsigned r = __brev((unsigned)e) >> 23;  // 9-bit reversal
    sre[r] = v.x;
    sim[r] = v.y;
  }
#else
  for (int e = t; e < 512; e += 256) {
    float2 v = buf[base + (long)e * stride];
    unsigned r = __brev((unsigned)e) >> 23;  // 9-bit reversal
    sre[r] = v.x;
    sim[r] = v.y;
  }
#endif
  __syncthreads();

  // 9 butterfly stages; each thread owns one disjoint (i0,i1) pair per stage
  for (int m = 2; m <= 512; m <<= 1) {
    int hf = m >> 1;
    int k = t & (hf - 1);
    int i0 = 2 * t - k;              // = (t/hf)*m + k
    int i1 = i0 + hf;
    float ang = sign * 6.28318530717958647692f * (float)k / (float)m;
    float s, c;
    __sincosf(ang, &s, &c);
    float xr = sre[i1], xi2 = sim[i1];
    float tr = c * xr - s * xi2;
    float ti = c * xi2 + s * xr;
    float ur = sre[i0], ui = sim[i0];
    sre[i0] = ur + tr; sim[i0] = ui + ti;
    sre[i1] = ur - tr; sim[i1] = ui - ti;
    __syncthreads();
  }

  // ---- scatter line back to global ----
#ifdef HAVE_ASYNC_LDS_ST
  for (int e = t; e < 512; e += 256) {
    float2 v; v.x = sre[e]; v.y = sim[e];
    stage[e] = v;
  }
  __syncthreads();
  for (int e = t; e < 512; e += 256) {
    __builtin_amdgcn_global_store_async_from_lds_b64(
        AS1P(buf + base + (long)e * stride), AS3P(&stage[e]), 0, 0);
  }
  __builtin_amdgcn_s_wait_asynccnt(0);
#else
  for (int e = t; e < 512; e += 256) {
    float2 v; v.x = sre[e]; v.y = sim[e];
    buf[base + (long)e * stride] = v;
  }
#endif
}

// ---------------------------------------------------------------------------
// 3) Per-pixel complex contraction on the matrix pipe:
//    Y[c,aout,p] = (1/65536) * sum_ain Fx[c,ain,p] * K[aout,ain,p]
//    One pixel per wave32. V_WMMA_F32_16X16X4_F32, M=c(12/16), N=aout(5/16),
//    K=ain(5) split into chunks of 4. Complex product = 4 real WMMAs/chunk;
//    f32 WMMA has no A/B negate (ISA NEG table), so -Im(A) is formed in VALU.
//    Operand loads are branchless: invalid lanes load plane 0 (L2-resident)
//    and are zeroed via cndmask, keeping EXEC all-1s (WMMA requirement) with
//    no saveexec regions. In-place safe: all reads of this pixel finish
//    in-register before the D-matrix stores.
// ---------------------------------------------------------------------------
__global__ void __launch_bounds__(256) wmma_contract_kernel(
    float2* __restrict__ buf,
    const float* __restrict__ kr, const float* __restrict__ ki) {
  int wave = threadIdx.x >> 5;
  int lane = threadIdx.x & 31;
  long p = (long)blockIdx.x * 8 + wave;   // pixel on padded grid
  int row = lane & 15;                    // A: M index (coil c); B: N index (aout)
  int kb = (lane >> 4) * 2;               // K sub-offset per 32-bit A/B layout

  // prefetch upcoming kernel-plane data (global_prefetch_b8 on gfx1250)
  __builtin_prefetch(kr + p + 4096, 0, 0);
  __builtin_prefetch(ki + p + 4096, 0, 0);

  v8f accR = {};
  v8f accI = {};
#pragma unroll
  for (int ch = 0; ch < 2; ++ch) {
    v2f Ar, Ai, nAi, Br, Bi;
#pragma unroll
    for (int v = 0; v < 2; ++v) {
      int ain = ch * 4 + kb + v;
      // A = Fx[c, ain, p]; clamp invalid lanes to plane 0, zero via cndmask
      bool aok = (row < C_DIM) && (ain < A_DIM);
      int aplane = aok ? (row * A_DIM + ain) : 0;
      float2 f = buf[((long)aplane << 18) + p];
      Ar[v] = aok ? f.x : 0.f;
      Ai[v] = aok ? f.y : 0.f;
      nAi[v] = aok ? -f.y : 0.f;
      // B = K[aout, ain, p]  (planar real/imag inputs)
      bool bok = (row < A_DIM) && (ain < A_DIM);
      int bplane = bok ? (row * A_DIM + ain) : 0;
      long kidx = ((long)bplane << 18) + p;
      float br = kr[kidx];
      float bi = ki[kidx];
      Br[v] = bok ? br : 0.f;
      Bi[v] = bok ? bi : 0.f;
    }
    // Yr += Ar*Br - Ai*Bi ; Yi += Ar*Bi + Ai*Br
    accR = __builtin_amdgcn_wmma_f32_16x16x4_f32(false, Ar, false, Br,
                                                 (short)0, accR, false, false);
    accR = __builtin_amdgcn_wmma_f32_16x16x4_f32(false, nAi, false, Bi,
                                                 (short)0, accR, false, false);
    accI = __builtin_amdgcn_wmma_f32_16x16x4_f32(false, Ar, false, Bi,
                                                 (short)0, accI, false, false);
    accI = __builtin_amdgcn_wmma_f32_16x16x4_f32(false, Ai, false, Br,
                                                 (short)0, accI, false, false);
  }

  // D layout: lane 0-15 -> N=lane, M=vgpr; lane 16-31 -> N=lane-16, M=vgpr+8
  int hi = lane >> 4;
  int n = lane & 15;
#pragma unroll
  for (int v = 0; v < 8; ++v) {
    int m = v + 8 * hi;
    if (m < C_DIM && n < A_DIM) {
      float2 o;
      o.x = accR[v] * TOTAL_SCALE;
      o.y = accI[v] * TOTAL_SCALE;
      buf[((long)(m * A_DIM + n) << 18) + p] = o;
    }
  }
}

// ---------------------------------------------------------------------------
// 4) centered crop + multiply by conj(mps) + sum over coils -> out[a,y,x]
// ---------------------------------------------------------------------------
__global__ void __launch_bounds__(256) crop_combine_kernel(
    const float2* __restrict__ buf,
    const float* __restrict__ mr, const float* __restrict__ mi,
    float2* __restrict__ out) {
  long i = (long)blockIdx.x * 256 + threadIdx.x;  // [a][y][x], 5*256*256
  int x = (int)(i & 255);
  int y = (int)((i >> 8) & 255);
  int a = (int)(i >> 16);
  long mbase = (long)y * W_DIM + x;
  long pbase = (long)(y + PAD) * WP + (x + PAD);
  float sr = 0.f, si = 0.f;
#pragma unroll 4
  for (int c = 0; c < C_DIM; ++c) {
    float2 yv = buf[((long)(c * A_DIM + a) << 18) + pbase];
    float m_r = mr[(long)c * (H_DIM * W_DIM) + mbase];
    float m_i = mi[(long)c * (H_DIM * W_DIM) + mbase];
    // y * conj(m)
    sr += yv.x * m_r + yv.y * m_i;
    si += yv.y * m_r - yv.x * m_i;
  }
  float2 o; o.x = sr; o.y = si;
  out[i] = o;
}

// ---------------------------------------------------------------------------
extern "C" void kernel_launch(void* const* d_in, const int* in_sizes, int n_in,
                              void* d_out, int out_size, void* d_ws,
                              size_t ws_size, hipStream_t stream) {
  (void)in_sizes; (void)n_in; (void)out_size; (void)ws_size;
  const float* xr = (const float*)d_in[0];
  const float* xi = (const float*)d_in[1];
  const float* mr = (const float*)d_in[2];
  const float* mi = (const float*)d_in[3];
  const float* kr = (const float*)d_in[4];
  const float* ki = (const float*)d_in[5];
  float2* buf = (float2*)d_ws;   // requires 60*512*512*8 = 125,829,120 bytes
  float2* out = (float2*)d_out;  // complex64 [5,256,256] interleaved

  // 1) modulate + centered pad into 60 planes of 512x512
  modpad_kernel<<<(NPLANES * NPIX) / 256, 256, 0, stream>>>(xr, xi, mr, mi, buf);
  // 2) forward 2-D FFT (rows then columns), unnormalized
  fft512_lines<<<NPLANES * HP, 256, 0, stream>>>(buf, 0, -1.0f);
  fft512_lines<<<NPLANES * WP, 256, 0, stream>>>(buf, 1, -1.0f);
  // 3) per-pixel [12x5]*[5x5] complex contraction on WMMA, in-place, scaled
  wmma_contract_kernel<<<NPIX / 8, 256, 0, stream>>>(buf, kr, ki);
  // 4) inverse 2-D FFT (columns then rows), scale already folded
  fft512_lines<<<NPLANES * WP, 256, 0, stream>>>(buf, 1, +1.0f);
  fft512_lines<<<NPLANES * HP, 256, 0, stream>>>(buf, 0, +1.0f);
  // 5) crop + conj(mps) coil reduction
  crop_combine_kernel<<<(A_DIM * H_DIM * W_DIM) / 256, 256, 0, stream>>>(
      buf, mr, mi, out);
}